// Subnetwork_43748536877075
// MI455X (gfx1250) — compile-verified
//
#include <hip/hip_runtime.h>

typedef __attribute__((ext_vector_type(2))) float v2f;
typedef __attribute__((ext_vector_type(8))) float v8f;

#define TPB_A 256
#define BLK_A 1024

// ---- tanh: gfx1250 hardware v_tanh_f32 (confirmed present) ----
#if __has_builtin(__builtin_amdgcn_tanhf)
#define TANHF(x) __builtin_amdgcn_tanhf(x)
#elif __has_builtin(__builtin_amdgcn_tanh_f32)
#define TANHF(x) __builtin_amdgcn_tanh_f32(x)
#else
#define TANHF(x) tanhf(x)
#endif

// ---- f32 WMMA (exact precision, 16x16x4) — confirmed to lower ----
#if __has_builtin(__builtin_amdgcn_wmma_f32_16x16x4_f32)
#define USE_WMMA 1
#endif

// Full per-sample computation in plain VALU (used for tails / fallback).
__device__ __forceinline__ void sample_valu(
    float xv,
    const float (&w1)[10], const float (&bb1)[10],
    const float* __restrict__ W2,
    const float (&bb2)[6], const float (&w3v)[6], float b3s,
    float& outv, float& g2o)
{
    float h1[10], t1[10], uu[10];
#pragma unroll
    for (int j = 0; j < 10; ++j) {
        float z = fmaf(xv, w1[j], bb1[j]);
        float h = TANHF(z);
        h1[j] = h;
        t1[j] = fmaf(-h, h, 1.0f);
        uu[j] = h * t1[j];
    }
    float z2v[6], dz2[6], d2z2[6];
#pragma unroll
    for (int m = 0; m < 6; ++m) { z2v[m] = 0.f; dz2[m] = 0.f; d2z2[m] = 0.f; }
#pragma unroll
    for (int k = 0; k < 10; ++k) {
        float w1k = w1[k];
        float ta = t1[k] * w1k;
        float tb = -2.0f * uu[k] * w1k * w1k;
#pragma unroll
        for (int m = 0; m < 6; ++m) {
            float w2 = W2[k * 6 + m];
            z2v[m]  = fmaf(h1[k], w2, z2v[m]);
            dz2[m]  = fmaf(ta,    w2, dz2[m]);
            d2z2[m] = fmaf(tb,    w2, d2z2[m]);
        }
    }
    outv = b3s; g2o = 0.f;
#pragma unroll
    for (int m = 0; m < 6; ++m) {
        float zz  = z2v[m] + bb2[m];
        float h2  = TANHF(zz);
        float t2  = fmaf(-h2, h2, 1.0f);
        float dh2 = t2 * dz2[m];
        float d2h2 = fmaf(-2.0f * h2 * dh2, dz2[m], t2 * d2z2[m]);
        outv = fmaf(h2,   w3v[m], outv);
        g2o  = fmaf(d2h2, w3v[m], g2o);
    }
}

__global__ __launch_bounds__(TPB_A) void mlp_fused(
    const float* __restrict__ x,
    const float* __restrict__ W1, const float* __restrict__ b1,
    const float* __restrict__ W2, const float* __restrict__ b2,
    const float* __restrict__ W3, const float* __restrict__ b3,
    float* __restrict__ outraw, float* __restrict__ ws, int n)
{
    const int tid    = blockIdx.x * TPB_A + threadIdx.x;
    const int stride = BLK_A * TPB_A;
    const int lane   = threadIdx.x & 31;
    const int l16    = lane & 15;
    const bool hiHalf = lane >= 16;

    // uniform weights -> SGPRs
    float w1[10], bb1[10];
#pragma unroll
    for (int j = 0; j < 10; ++j) { w1[j] = W1[j]; bb1[j] = b1[j]; }
    float bb2[6], w3v[6];
#pragma unroll
    for (int m = 0; m < 6; ++m) { bb2[m] = b2[m]; w3v[m] = W3[m]; }
    const float b3s = b3[0];

#ifdef USE_WMMA
    // Constant A tiles: rows 0..5 = W^T, rows 6..15 = 0 (so C rows 6..15 are 0
    // and padded K slots in B are multiplied by zero).
    // A layout (16x4 f32): lanes 0-15: v0=K(4c), v1=K(4c+1); lanes 16-31: v0=K(4c+2), v1=K(4c+3).
    v2f AW[3], AP[3], APP[3];
    // Per-lane layer-1 constants in B-layout: slot (c,e) handles K = 4c + (hi?2:0) + e.
    float w1B[6], b1B[6];
    {
        float w2c[10];
#pragma unroll
        for (int k = 0; k < 10; ++k) w2c[k] = (l16 < 6) ? W2[k * 6 + l16] : 0.0f;
#pragma unroll
        for (int c = 0; c < 3; ++c) {
            const int kl0 = 4 * c, kl1 = 4 * c + 1, kh0 = 4 * c + 2, kh1 = 4 * c + 3;
            float wl0 = w2c[kl0], wl1 = w2c[kl1];
            float wh0 = (kh0 < 10) ? w2c[kh0] : 0.0f;
            float wh1 = (kh1 < 10) ? w2c[kh1] : 0.0f;
            float s0  = hiHalf ? wh0 : wl0;
            float s1e = hiHalf ? wh1 : wl1;
            float q0 = hiHalf ? ((kh0 < 10) ? w1[kh0] : 0.0f) : w1[kl0];
            float q1 = hiHalf ? ((kh1 < 10) ? w1[kh1] : 0.0f) : w1[kl1];
            AW[c].x  = s0;                     AW[c].y  = s1e;                    // W2
            AP[c].x  = q0 * s0;                AP[c].y  = q1 * s1e;               // W1*W2
            APP[c].x = -2.0f * q0 * q0 * s0;   APP[c].y = -2.0f * q1 * q1 * s1e;  // -2*W1^2*W2
            w1B[2 * c + 0] = q0;
            w1B[2 * c + 1] = q1;
            float p0 = hiHalf ? ((kh0 < 10) ? bb1[kh0] : 0.0f) : bb1[kl0];
            float p1 = hiHalf ? ((kh1 < 10) ? bb1[kh1] : 0.0f) : bb1[kl1];
            b1B[2 * c + 0] = p0;
            b1B[2 * c + 1] = p1;
        }
    }
#endif

    float s1 = 0.f, s2 = 0.f, s3 = 0.f;
    const int iters = n / stride;

    for (int it = 0; it < iters; ++it) {
        const int i = tid + it * stride;
        __builtin_prefetch(x + i + stride, 0, 1);   // global_prefetch_b8
        const float xv = x[i];
        float outv, g2;
#ifdef USE_WMMA
        // Build B tiles directly in B-layout from the broadcast x of each
        // sample: only ONE ds_bpermute per 16-sample tile; layer 1
        // (fma + v_tanh) is evaluated in place.
        v8f aZ[2], aD[2], aDD[2];
#pragma unroll
        for (int t = 0; t < 2; ++t) {
            const int src = t * 16 + l16;
            const float xs = __shfl(xv, src);
            v8f cz  = {0.f, 0.f, 0.f, 0.f, 0.f, 0.f, 0.f, 0.f};
            v8f cd  = {0.f, 0.f, 0.f, 0.f, 0.f, 0.f, 0.f, 0.f};
            v8f cdd = {0.f, 0.f, 0.f, 0.f, 0.f, 0.f, 0.f, 0.f};
#pragma unroll
            for (int c = 0; c < 3; ++c) {
                float h0  = TANHF(fmaf(xs, w1B[2 * c + 0], b1B[2 * c + 0]));
                float h1e = TANHF(fmaf(xs, w1B[2 * c + 1], b1B[2 * c + 1]));
                float t0  = fmaf(-h0,  h0,  1.0f);
                float t1e = fmaf(-h1e, h1e, 1.0f);
                v2f Bh; Bh.x = h0;       Bh.y = h1e;        // h1
                v2f Bt; Bt.x = t0;       Bt.y = t1e;        // 1-h1^2
                v2f Bu; Bu.x = h0 * t0;  Bu.y = h1e * t1e;  // h1*(1-h1^2)
                cz  = __builtin_amdgcn_wmma_f32_16x16x4_f32(false, AW[c],  false, Bh, (short)0, cz,  false, false);
                cd  = __builtin_amdgcn_wmma_f32_16x16x4_f32(false, AP[c],  false, Bt, (short)0, cd,  false, false);
                cdd = __builtin_amdgcn_wmma_f32_16x16x4_f32(false, APP[c], false, Bu, (short)0, cdd, false, false);
            }
            aZ[t] = cz; aD[t] = cd; aDD[t] = cdd;
        }
        // C layout: register m, lanes 0-15 hold row m (z2[m] etc.) for the
        // tile's 16 samples. Lanes 16-31 fetch tile-1 results from lane-16.
        outv = b3s; g2 = 0.f;
#pragma unroll
        for (int m = 0; m < 6; ++m) {
            float sZ  = __shfl(aZ[1][m],  l16);
            float sD  = __shfl(aD[1][m],  l16);
            float sDD = __shfl(aDD[1][m], l16);
            float z2e   = hiHalf ? sZ  : aZ[0][m];
            float dz2e  = hiHalf ? sD  : aD[0][m];
            float d2z2e = hiHalf ? sDD : aDD[0][m];
            float zz  = z2e + bb2[m];
            float h2  = TANHF(zz);
            float t2  = fmaf(-h2, h2, 1.0f);
            float dh2 = t2 * dz2e;
            float d2h2 = fmaf(-2.0f * h2 * dh2, dz2e, t2 * d2z2e);
            outv = fmaf(h2,   w3v[m], outv);
            g2   = fmaf(d2h2, w3v[m], g2);
        }
#else
        sample_valu(xv, w1, bb1, W2, bb2, w3v, b3s, outv, g2);
#endif
        outraw[i] = outv;
        s1 += outv; s2 += outv * outv; s3 += g2 * g2;
    }

    // generic tail (empty for N = 4M with this grid) — no WMMA (needs full EXEC)
    for (int i = iters * stride + tid; i < n; i += stride) {
        float outv, g2;
        sample_valu(x[i], w1, bb1, W2, bb2, w3v, b3s, outv, g2);
        outraw[i] = outv;
        s1 += outv; s2 += outv * outv; s3 += g2 * g2;
    }

    // deterministic wave reduction (fixed xor order)
#pragma unroll
    for (int off = 16; off > 0; off >>= 1) {
        s1 += __shfl_xor(s1, off);
        s2 += __shfl_xor(s2, off);
        s3 += __shfl_xor(s3, off);
    }
    __shared__ float red[3][TPB_A / 32];
    const int wid = threadIdx.x >> 5;
    if (lane == 0) { red[0][wid] = s1; red[1][wid] = s2; red[2][wid] = s3; }
    __syncthreads();
    if (threadIdx.x == 0) {
        float a = 0.f, bsum = 0.f, c = 0.f;
#pragma unroll
        for (int w = 0; w < TPB_A / 32; ++w) { a += red[0][w]; bsum += red[1][w]; c += red[2][w]; }
        ws[16 + 3 * blockIdx.x + 0] = a;
        ws[16 + 3 * blockIdx.x + 1] = bsum;
        ws[16 + 3 * blockIdx.x + 2] = c;
    }
}

__global__ __launch_bounds__(256) void reduce_finalize(float* __restrict__ ws,
                                                       float* __restrict__ pen_out,
                                                       float n_f)
{
    __shared__ float sh[3][256];
    float a = 0.f, b = 0.f, c = 0.f;
    for (int t = threadIdx.x; t < BLK_A; t += 256) {
        a += ws[16 + 3 * t + 0];
        b += ws[16 + 3 * t + 1];
        c += ws[16 + 3 * t + 2];
    }
    sh[0][threadIdx.x] = a; sh[1][threadIdx.x] = b; sh[2][threadIdx.x] = c;
    __syncthreads();
    for (int s = 128; s > 0; s >>= 1) {
        if (threadIdx.x < s) {
            sh[0][threadIdx.x] += sh[0][threadIdx.x + s];
            sh[1][threadIdx.x] += sh[1][threadIdx.x + s];
            sh[2][threadIdx.x] += sh[2][threadIdx.x + s];
        }
        __syncthreads();
    }
    if (threadIdx.x == 0) {
        float S1 = sh[0][0], S2 = sh[1][0], S3 = sh[2][0];
        float mean = S1 / n_f;
        float var  = fmaxf(S2 / n_f - mean * mean, 0.0f);
        float nrm  = fmaxf(sqrtf(var), 1e-10f);
        ws[0] = mean;
        ws[1] = 1.0f / nrm;
        *pen_out = (S3 / n_f) / nrm;
    }
}

__global__ __launch_bounds__(256) void normalize_k(float* __restrict__ out,
                                                   const float* __restrict__ ws, int n)
{
    const float mean = ws[0];
    const float inv  = ws[1];
    int i = (blockIdx.x * 256 + threadIdx.x) * 4;
    if (i + 3 < n) {
        float4 v = *(float4*)(out + i);
        v.x = (v.x - mean) * inv;
        v.y = (v.y - mean) * inv;
        v.z = (v.z - mean) * inv;
        v.w = (v.w - mean) * inv;
        *(float4*)(out + i) = v;
    } else {
        for (; i < n; ++i) out[i] = (out[i] - mean) * inv;
    }
}

extern "C" void kernel_launch(void* const* d_in, const int* in_sizes, int n_in,
                              void* d_out, int out_size, void* d_ws, size_t ws_size,
                              hipStream_t stream) {
    const float* x  = (const float*)d_in[0];
    const float* W1 = (const float*)d_in[1];
    const float* b1 = (const float*)d_in[2];
    const float* W2 = (const float*)d_in[3];
    const float* b2 = (const float*)d_in[4];
    const float* W3 = (const float*)d_in[5];
    const float* b3 = (const float*)d_in[6];
    const int n = in_sizes[0];
    float* out = (float*)d_out;
    float* ws  = (float*)d_ws;

    mlp_fused<<<BLK_A, TPB_A, 0, stream>>>(x, W1, b1, W2, b2, W3, b3, out, ws, n);
    reduce_finalize<<<1, 256, 0, stream>>>(ws, out + n, (float)n);
    int blocksC = (n + 1023) / 1024;
    normalize_k<<<blocksC, 256, 0, stream>>>(out, ws, n);
}